// TwoLayerGCN_36249523978361
// MI455X (gfx1250) — compile-verified
//
#include <hip/hip_runtime.h>

#define N_NODES 100000
#define N_EDGES 1600000
#define IN_F    256
#define HID_F   128
#define OUT_F   32

typedef float v2f __attribute__((ext_vector_type(2)));
typedef float v8f __attribute__((ext_vector_type(8)));

// ---------------------------------------------------------------------------
// GEMM1: h0[N,128] = x[N,256] @ W1[256,128], f32 via V_WMMA_F32_16X16X4_F32.
// One wave computes a 16-row block across all 8 column tiles (16x128 output).
// A frag (16x4): lane l (m = l&15, kh = l>>4) holds K = 2*kh + {0,1} of row m.
// B frag (4x16): lane l holds W1[K = 2*kh + {0,1}][n0 + m].
// C/D (16x16):   lane l, vgpr v holds D[kh*8 + v][m].
// ---------------------------------------------------------------------------
__global__ __launch_bounds__(128) void gemm1_wmma(
    const float* __restrict__ x, const float* __restrict__ W1,
    float* __restrict__ h0)
{
    const int wid  = (blockIdx.x * blockDim.x + threadIdx.x) >> 5;
    const int lane = threadIdx.x & 31;
    if (wid >= N_NODES / 16) return;          // wave-uniform: EXEC stays full
    const int m    = lane & 15;
    const int kh   = lane >> 4;
    const int row0 = wid * 16;

    v8f acc[8] = {};

    const float* arow = x + (size_t)(row0 + m) * IN_F + 2 * kh;
    const float* wcol = W1 + m;

    for (int k = 0; k < IN_F; k += 4) {
        v2f a = *(const v2f*)(arow + k);      // contiguous K-pair, b64 load
        const float* wrow = wcol + (size_t)(k + 2 * kh) * HID_F;
#pragma unroll
        for (int t = 0; t < 8; ++t) {
            v2f b;
            b.x = wrow[t * 16];
            b.y = wrow[HID_F + t * 16];
            acc[t] = __builtin_amdgcn_wmma_f32_16x16x4_f32(
                false, a, false, b, (short)0, acc[t], false, false);
        }
    }

    float* drow = h0 + (size_t)(row0 + kh * 8) * HID_F + m;
#pragma unroll
    for (int t = 0; t < 8; ++t)
#pragma unroll
        for (int v = 0; v < 8; ++v)
            drow[(size_t)v * HID_F + t * 16] = acc[t][v];
}

// ---------------------------------------------------------------------------
// GEMM2: h2[N,32] = relu(h1)[N,128] @ W2[128,32]; ReLU fused into A-load.
// ---------------------------------------------------------------------------
__global__ __launch_bounds__(128) void gemm2_wmma(
    const float* __restrict__ h1, const float* __restrict__ W2,
    float* __restrict__ h2)
{
    const int wid  = (blockIdx.x * blockDim.x + threadIdx.x) >> 5;
    const int lane = threadIdx.x & 31;
    if (wid >= N_NODES / 16) return;
    const int m    = lane & 15;
    const int kh   = lane >> 4;
    const int row0 = wid * 16;

    v8f acc[2] = {};

    const float* arow = h1 + (size_t)(row0 + m) * HID_F + 2 * kh;
    const float* wcol = W2 + m;

    for (int k = 0; k < HID_F; k += 4) {
        v2f a = *(const v2f*)(arow + k);
        a.x = fmaxf(a.x, 0.0f);               // fused ReLU
        a.y = fmaxf(a.y, 0.0f);
        const float* wrow = wcol + (size_t)(k + 2 * kh) * OUT_F;
#pragma unroll
        for (int t = 0; t < 2; ++t) {
            v2f b;
            b.x = wrow[t * 16];
            b.y = wrow[OUT_F + t * 16];
            acc[t] = __builtin_amdgcn_wmma_f32_16x16x4_f32(
                false, a, false, b, (short)0, acc[t], false, false);
        }
    }

    float* drow = h2 + (size_t)(row0 + kh * 8) * OUT_F + m;
#pragma unroll
    for (int t = 0; t < 2; ++t)
#pragma unroll
        for (int v = 0; v < 8; ++v)
            drow[(size_t)v * OUT_F + t * 16] = acc[t][v];
}

// ---------------------------------------------------------------------------
// SpMM, F=128: one wave per edge; 32 lanes x float4 = one full row.
// Gather hits L2 (h0 = 51 MB < 192 MB L2); scatter via f32 atomics (no-return
// -> STOREcnt path, resolved at L2).
// ---------------------------------------------------------------------------
__global__ __launch_bounds__(256) void spmm_f128(
    const int* __restrict__ rows, const int* __restrict__ cols,
    const float* __restrict__ vals, const float* __restrict__ src,
    float* __restrict__ dst)
{
    const int e    = (blockIdx.x * blockDim.x + threadIdx.x) >> 5;
    const int lane = threadIdx.x & 31;
    if (e >= N_EDGES) return;
    const int   r = rows[e];
    const int   c = cols[e];
    const float w = vals[e];
    const float4 mv = ((const float4*)(src + (size_t)c * HID_F))[lane];
    float* d = dst + (size_t)r * HID_F + lane * 4;
    __hip_atomic_fetch_add(d + 0, w * mv.x, __ATOMIC_RELAXED, __HIP_MEMORY_SCOPE_AGENT);
    __hip_atomic_fetch_add(d + 1, w * mv.y, __ATOMIC_RELAXED, __HIP_MEMORY_SCOPE_AGENT);
    __hip_atomic_fetch_add(d + 2, w * mv.z, __ATOMIC_RELAXED, __HIP_MEMORY_SCOPE_AGENT);
    __hip_atomic_fetch_add(d + 3, w * mv.w, __ATOMIC_RELAXED, __HIP_MEMORY_SCOPE_AGENT);
}

// SpMM, F=32: one wave per edge; one lane per feature.
__global__ __launch_bounds__(256) void spmm_f32(
    const int* __restrict__ rows, const int* __restrict__ cols,
    const float* __restrict__ vals, const float* __restrict__ src,
    float* __restrict__ dst)
{
    const int e    = (blockIdx.x * blockDim.x + threadIdx.x) >> 5;
    const int lane = threadIdx.x & 31;
    if (e >= N_EDGES) return;
    const int   r = rows[e];
    const int   c = cols[e];
    const float w = vals[e];
    const float mv = src[(size_t)c * OUT_F + lane];
    __hip_atomic_fetch_add(dst + (size_t)r * OUT_F + lane, w * mv,
                           __ATOMIC_RELAXED, __HIP_MEMORY_SCOPE_AGENT);
}

__global__ __launch_bounds__(256) void zero_f4(float4* __restrict__ p, int n4)
{
    int i = blockIdx.x * blockDim.x + threadIdx.x;
    if (i < n4) p[i] = make_float4(0.f, 0.f, 0.f, 0.f);
}

extern "C" void kernel_launch(void* const* d_in, const int* in_sizes, int n_in,
                              void* d_out, int out_size, void* d_ws, size_t ws_size,
                              hipStream_t stream)
{
    const float* x    = (const float*)d_in[0];
    const int*   rows = (const int*)  d_in[1];
    const int*   cols = (const int*)  d_in[2];
    const float* vals = (const float*)d_in[3];
    const float* W1   = (const float*)d_in[4];
    const float* W2   = (const float*)d_in[5];
    float* out = (float*)d_out;

    // Workspace layout: buf0 = h0 (then reused as h2), buf1 = h1.
    float* h0 = (float*)d_ws;                       // N*128 f32
    float* h1 = h0 + (size_t)N_NODES * HID_F;       // N*128 f32
    float* h2 = h0;                                 // h0 dead after spmm1

    const int gemmBlocks = ((N_NODES / 16) + 3) / 4;        // 4 waves/block
    const int edgeBlocks = (N_EDGES * 32 + 255) / 256;      // 1 wave/edge

    // 1) h0 = x @ W1
    gemm1_wmma<<<gemmBlocks, 128, 0, stream>>>(x, W1, h0);

    // 2) h1 = 0 ; h1 += A * h0
    {
        int n4 = (N_NODES * HID_F) / 4;
        zero_f4<<<(n4 + 255) / 256, 256, 0, stream>>>((float4*)h1, n4);
    }
    spmm_f128<<<edgeBlocks, 256, 0, stream>>>(rows, cols, vals, h0, h1);

    // 3) h2 = relu(h1) @ W2
    gemm2_wmma<<<gemmBlocks, 128, 0, stream>>>(h1, W2, h2);

    // 4) out = 0 ; out += A * h2
    {
        int n4 = (N_NODES * OUT_F) / 4;
        zero_f4<<<(n4 + 255) / 256, 256, 0, stream>>>((float4*)out, n4);
    }
    spmm_f32<<<edgeBlocks, 256, 0, stream>>>(rows, cols, vals, h2, out);
}